// CANN3DNet_41042707481170
// MI455X (gfx1250) — compile-verified
//
#include <hip/hip_runtime.h>
#include <math.h>

// ---------------------------------------------------------------------------
// CANN3DNet on MI455X (gfx1250): batch-1 GEMV chain, HBM-bound (~277 MB
// of fp32 weights => ~12 us at 23.3 TB/s). fp32 WMMA 16x16x4 keeps exact
// precision while engaging the matrix pipe (confirmed lowering last round).
// This revision templates N/KPS so all weight-row strides become immediate
// offsets in global_load (no per-iteration address math), uses 4 independent
// WMMA accumulator chains, and removes the dead remainder path.
// Dead f-gate columns and Wr1/Wr2 (multiplied by zero state) are skipped.
// ---------------------------------------------------------------------------

typedef __attribute__((ext_vector_type(2))) float v2f;
typedef __attribute__((ext_vector_type(8))) float v8f;

#define Hdim   3072
#define H4dim  12288
#define NUMSd  1024
#define DIN    6

__device__ __forceinline__ float sigmoidf_(float v) { return 1.0f / (1.0f + expf(-v)); }

// ---- stage 0: t0 = relu(x @ W_in0), x is 1x6 -------------------------------
__global__ void k_in0(const float* __restrict__ x, const float* __restrict__ W0,
                      float* __restrict__ t0) {
    int j = blockIdx.x * blockDim.x + threadIdx.x;
    if (j >= Hdim) return;
    float s = 0.f;
#pragma unroll
    for (int k = 0; k < DIN; ++k) s = fmaf(x[k], W0[k * Hdim + j], s);
    t0[j] = s > 0.f ? s : 0.f;
}

// ---- generic split-K GEMV via V_WMMA_F32_16X16X4_F32 -----------------------
// One wave owns one 16-column tile over one K-chunk of length KPS.
//   A (16x4 f32): row 0 = x[k..k+3], rows 1..15 = 0.
//     lane L holds M=L%16; lanes 0-15: K=0,1 in V0,V1; lanes 16-31: K=2,3.
//   B (4x16 f32): lane L holds column N=L%16; lanes 0-15: K=0,1; 16-31: K=2,3.
//   C/D: VGPR0 lanes 0-15 = row M=0 (the GEMV result).
// REMAP != 0 skips the dead f-gate block: compact col c -> c (c<REMAP) else c+REMAP.
// Compile-time N => all 13 row strides fold into 24-bit immediate offsets.
template <int N, int KPS, int REMAP>
__global__ void __launch_bounds__(128) k_gemv_wmma(
    const float* __restrict__ W, const float* __restrict__ x,
    float* __restrict__ partial) {
    constexpr int CN = N - REMAP;       // compact (stored) column count
    const int lane = threadIdx.x;       // 0..31, one wave per threadIdx.y
    const int half = lane >> 4;         // 0 or 1
    const int l15  = lane & 15;
    const int tile = blockIdx.x * blockDim.y + threadIdx.y;
    const int j0c  = tile * 16;
    const int j0   = (REMAP != 0 && j0c >= REMAP) ? (j0c + REMAP) : j0c;
    const int kBeg = blockIdx.y * KPS;
    const int xoff = half * 2;

    const float* __restrict__ Wp = W + (size_t)(kBeg + xoff) * N + j0 + l15;
    const float* __restrict__ xp = x + kBeg + xoff;

#if __has_builtin(__builtin_amdgcn_wmma_f32_16x16x4_f32)
    v8f acc0 = {0.f, 0.f, 0.f, 0.f, 0.f, 0.f, 0.f, 0.f};
    v8f acc1 = acc0, acc2 = acc0, acc3 = acc0;  // 4 independent RAW chains
#pragma unroll 2
    for (int it = 0; it < KPS / 16; ++it) {
        v2f b0, b1, b2, b3;
        b0.x = Wp[(size_t)0  * N]; b0.y = Wp[(size_t)1  * N];
        b1.x = Wp[(size_t)4  * N]; b1.y = Wp[(size_t)5  * N];
        b2.x = Wp[(size_t)8  * N]; b2.y = Wp[(size_t)9  * N];
        b3.x = Wp[(size_t)12 * N]; b3.y = Wp[(size_t)13 * N];
        v2f a0 = {0.f, 0.f}, a1 = a0, a2 = a0, a3 = a0;
        if (l15 == 0) {  // only lanes 0 and 16 carry A row 0
            a0.x = xp[0];  a0.y = xp[1];
            a1.x = xp[4];  a1.y = xp[5];
            a2.x = xp[8];  a2.y = xp[9];
            a3.x = xp[12]; a3.y = xp[13];
        }
        acc0 = __builtin_amdgcn_wmma_f32_16x16x4_f32(false, a0, false, b0,
                                                     (short)0, acc0, false, false);
        acc1 = __builtin_amdgcn_wmma_f32_16x16x4_f32(false, a1, false, b1,
                                                     (short)0, acc1, false, false);
        acc2 = __builtin_amdgcn_wmma_f32_16x16x4_f32(false, a2, false, b2,
                                                     (short)0, acc2, false, false);
        acc3 = __builtin_amdgcn_wmma_f32_16x16x4_f32(false, a3, false, b3,
                                                     (short)0, acc3, false, false);
        Wp += (size_t)16 * N;
        xp += 16;
    }
    if (half == 0)
        partial[(size_t)blockIdx.y * CN + j0c + l15] =
            (acc0[0] + acc1[0]) + (acc2[0] + acc3[0]);
#else
    // Fallback: plain FMA GEMV, lane halves split K parity, combine via shuffle.
    float s = 0.f;
    const float* __restrict__ Wf = W + (size_t)j0 + l15;
    for (int k = kBeg + half; k < kBeg + KPS; k += 2)
        s = fmaf(x[k], Wf[(size_t)k * N], s);
    s += __shfl_xor(s, 16, 32);
    if (half == 0)
        partial[(size_t)blockIdx.y * CN + j0c + l15] = s;
#endif
}

// ---- reducers --------------------------------------------------------------
__global__ void k_reduce_copy(const float* __restrict__ P, float* __restrict__ out0,
                              float* __restrict__ out1, int N, int S) {
    int j = blockIdx.x * blockDim.x + threadIdx.x;
    if (j >= N) return;
    float s = 0.f;
    for (int i = 0; i < S; ++i) s += P[(size_t)i * N + j];
    out0[j] = s;
    if (out1) out1[j] = s;
}

// h = sigmoid(z_o) * tanh(sigmoid(z_i) * tanh(z_g));  f-gate dead (c0 == 0).
__global__ void k_reduce_lstm(const float* __restrict__ P, float* __restrict__ h,
                              float* __restrict__ hcopy, int S) {
    int n = blockIdx.x * blockDim.x + threadIdx.x;
    if (n >= Hdim) return;
    const int cN = 3 * Hdim;  // compact columns: [i | g | o]
    float zi = 0.f, zg = 0.f, zo = 0.f;
    for (int i = 0; i < S; ++i) {
        const float* row = P + (size_t)i * cN;
        zi += row[n];
        zg += row[Hdim + n];
        zo += row[2 * Hdim + n];
    }
    float iv = sigmoidf_(zi);
    float gv = tanhf(zg);
    float ov = sigmoidf_(zo);
    float hv = ov * tanhf(iv * gv);
    h[n] = hv;
    if (hcopy) hcopy[n] = hv;
}

__global__ void k_reduce_bias_relu(const float* __restrict__ P, const float* __restrict__ b,
                                   float* __restrict__ u, int N, int S) {
    int j = blockIdx.x * blockDim.x + threadIdx.x;
    if (j >= N) return;
    float s = 0.f;
    for (int i = 0; i < S; ++i) s += P[(size_t)i * N + j];
    s += b[j];
    u[j] = s > 0.f ? s : 0.f;
}

// ---- final 1024 -> 6 -------------------------------------------------------
__global__ void k_final(const float* __restrict__ u, const float* __restrict__ W,
                        const float* __restrict__ b, float* __restrict__ out) {
    __shared__ float red[128];
    int col = blockIdx.x, t = threadIdx.x;
    float s = 0.f;
    for (int k = t; k < NUMSd; k += 128) s = fmaf(u[k], W[k * 6 + col], s);
    red[t] = s;
    __syncthreads();
    for (int w = 64; w > 0; w >>= 1) {
        if (t < w) red[t] += red[t + w];
        __syncthreads();
    }
    if (t == 0) out[col] = red[0] + b[col];
}

// ---------------------------------------------------------------------------
extern "C" void kernel_launch(void* const* d_in, const int* in_sizes, int n_in,
                              void* d_out, int out_size, void* d_ws, size_t ws_size,
                              hipStream_t stream) {
    (void)in_sizes; (void)n_in; (void)out_size; (void)ws_size;
    const float* x    = (const float*)d_in[0];
    const float* Win0 = (const float*)d_in[1];
    const float* Win1 = (const float*)d_in[2];
    const float* Wk1  = (const float*)d_in[3];
    /* d_in[4] = Wr1 : unused (h0 == 0) */
    const float* Wk2  = (const float*)d_in[5];
    /* d_in[6] = Wr2 : unused (h0 == 0) */
    const float* Wd1  = (const float*)d_in[7];
    const float* bd1  = (const float*)d_in[8];
    const float* Wd2  = (const float*)d_in[9];
    const float* bd2  = (const float*)d_in[10];

    float* out = (float*)d_out;           // [input_out(3072) | state_out(3072) | dec(6)]
    float* ws  = (float*)d_ws;
    float* t0 = ws;                        // H
    float* xt = ws + Hdim;                 // H
    float* h1 = ws + 2 * Hdim;             // H
    float* h2 = ws + 3 * Hdim;             // H
    float* u  = ws + 4 * Hdim;             // NUMS
    float* P  = ws + 4 * Hdim + NUMSd;     // split-K partials, max 8*9216 floats

    dim3 blk(32, 4);  // 4 waves/block, each owns a 16-col tile

    // t0 = relu(x @ W_in0)
    k_in0<<<(Hdim + 255) / 256, 256, 0, stream>>>(x, Win0, t0);

    // input_out = t0 @ W_in1   (K=3072, N=3072, S=16, KPS=192)
    k_gemv_wmma<3072, 192, 0><<<dim3(48, 16), blk, 0, stream>>>(Win1, t0, P);
    k_reduce_copy<<<(Hdim + 255) / 256, 256, 0, stream>>>(P, out, xt, Hdim, 16);

    // LSTM1: z = xt @ Wk1, i/g/o columns only (K=3072, CN=9216, S=8, KPS=384)
    k_gemv_wmma<12288, 384, 3072><<<dim3(144, 8), blk, 0, stream>>>(Wk1, xt, P);
    k_reduce_lstm<<<(Hdim + 255) / 256, 256, 0, stream>>>(P, h1, nullptr, 8);

    // LSTM2: z = h1 @ Wk2 (same structure); h2 also goes to d_out[3072:6144]
    k_gemv_wmma<12288, 384, 3072><<<dim3(144, 8), blk, 0, stream>>>(Wk2, h1, P);
    k_reduce_lstm<<<(Hdim + 255) / 256, 256, 0, stream>>>(P, h2, out + Hdim, 8);

    // u = relu(h2 @ Wd1 + bd1)   (K=3072, N=1024, S=16, KPS=192)
    k_gemv_wmma<1024, 192, 0><<<dim3(16, 16), blk, 0, stream>>>(Wd1, h2, P);
    k_reduce_bias_relu<<<(NUMSd + 255) / 256, 256, 0, stream>>>(P, bd1, u, NUMSd, 16);

    // decoder_out = u @ Wd2 + bd2
    k_final<<<6, 128, 0, stream>>>(u, Wd2, bd2, out + 2 * Hdim);
}